// PINOBlock_26594437497671
// MI455X (gfx1250) — compile-verified
//
#include <hip/hip_runtime.h>
#include <math.h>

// PINO spectral-attention block for MI455X (gfx1250, wave32, WMMA).
// All matrix contractions run on V_WMMA_F32_16X16X4_F32 (full fp32 accuracy).
// Round 2: loop-invariant B-fragments (a_modes^T, x_modes rows, fc_w^T, bias)
// hoisted into registers to cut per-tile LDS traffic and VALU address math.

#define N_HEADS 4
#define N_MODES 32
#define C_IN    32
#define C_OUT   32
#define NBATCH  2
#define NPTS    110592              // 48*48*48 points per batch
#define TWO_PI  6.28318530717958647692f

typedef float v2f __attribute__((ext_vector_type(2)));
typedef float v8f __attribute__((ext_vector_type(8)));

// D(16x16 f32) = A(16x4 f32) x B(4x16 f32) + C
// Fragment layout (wave32):
//   A: lane m=lane&15 holds M=m, K = (lane>>4)*2 + {0,1} within each K=4 slab
//   B: lane n=lane&15 holds N=n, K = (lane>>4)*2 + {0,1}
//   C/D: VGPR v holds M = v + 8*(lane>>4), N = lane&15
__device__ __forceinline__ v8f wmma_f32_k4(v2f a, v2f b, v8f c) {
  return __builtin_amdgcn_wmma_f32_16x16x4_f32(false, a, false, b, (short)0, c,
                                               false, false);
}

// ---------------------------------------------------------------------------
// Kernel 1: AX[b,h,f,g] = sum_p A[p,f] * exp(-2pi*i * x[p]·x_modes[h,g])
// Each wave owns one head; each block owns a chunk of points of one batch.
// Partial 32x32 complex AX flushed with fp32 HW atomics (64KB, L2-resident).
// ---------------------------------------------------------------------------
#define K1_BLOCKS_PER_BATCH 108
#define K1_TILES            64      // 16-point tiles per block -> 1024 pts

__global__ void __launch_bounds__(128)
k1_accumulate_AX(const float* __restrict__ a,
                 const float* __restrict__ x,
                 const float* __restrict__ a_modes,
                 const float* __restrict__ x_modes,
                 float* __restrict__ axbuf)
{
  extern __shared__ float smem[];
  float* amT = smem;                       // [4][32][32]: amT[h][i][f]
  float* xm  = amT + N_HEADS * C_IN * N_MODES;   // [4][32][3]

  const int tid  = threadIdx.x;
  const int wave = tid >> 5;
  const int lane = tid & 31;
  const int m    = lane & 15;
  const int hi   = lane >> 4;

  float* wv = xm + N_HEADS * N_MODES * 3 + wave * (512 + 64 + 512);
  float* aS = wv;                          // [16][32] input tile
  float* xS = aS + 512;                    // [16][4]  coords
  float* AS = xS + 64;                     // [16][32] projected A tile

  // Stage a_modes transposed (B-fragment friendly) and x_modes.
  for (int idx = tid; idx < N_HEADS * N_MODES * C_IN; idx += blockDim.x) {
    int h = idx >> 10, r = idx & 1023, f = r >> 5, i = r & 31;
    amT[(h << 10) + i * N_MODES + f] = a_modes[idx];
  }
  for (int idx = tid; idx < N_HEADS * N_MODES * 3; idx += blockDim.x)
    xm[idx] = x_modes[idx];
  __syncthreads();

  const int h     = wave;                  // one head per wave
  const int batch = blockIdx.x / K1_BLOCKS_PER_BATCH;
  const int blk   = blockIdx.x % K1_BLOCKS_PER_BATCH;

  // ---- Hoist loop-invariant fragments into registers --------------------
  v2f amf[8][2];                           // a_modes^T B-fragments
  #pragma unroll
  for (int ks = 0; ks < 8; ++ks) {
    const int kb = ks * 4 + hi * 2;
    #pragma unroll
    for (int nt = 0; nt < 2; ++nt) {
      amf[ks][nt][0] = amT[(h << 10) + kb * N_MODES + nt * 16 + m];
      amf[ks][nt][1] = amT[(h << 10) + (kb + 1) * N_MODES + nt * 16 + m];
    }
  }
  float xmg[2][3];                         // this lane's x_modes rows (g=gt*16+m)
  #pragma unroll
  for (int gt = 0; gt < 2; ++gt)
    #pragma unroll
    for (int d = 0; d < 3; ++d)
      xmg[gt][d] = xm[h * (N_MODES * 3) + (gt * 16 + m) * 3 + d];

  const v8f vz = {0.f,0.f,0.f,0.f,0.f,0.f,0.f,0.f};
  v8f accRe[2][2], accIm[2][2];
  #pragma unroll
  for (int i = 0; i < 2; ++i)
    #pragma unroll
    for (int j = 0; j < 2; ++j) { accRe[i][j] = vz; accIm[i][j] = vz; }

  for (int t = 0; t < K1_TILES; ++t) {
    const long p0 = (long)batch * NPTS + (long)blk * (K1_TILES * 16) + t * 16;

    for (int r = 0; r < 16; ++r)                       // coalesced 128B rows
      aS[r * 32 + lane] = a[(p0 + r) * C_IN + lane];
    if (lane < 16) {
      xS[lane * 4 + 0] = x[(p0 + lane) * 3 + 0];
      xS[lane * 4 + 1] = x[(p0 + lane) * 3 + 1];
      xS[lane * 4 + 2] = x[(p0 + lane) * 3 + 2];
    }

    // --- A(16x32) = aS(16x32) x amT[h](32x32) via 16 WMMAs -----------------
    v8f pa[2] = {vz, vz};
    #pragma unroll
    for (int ks = 0; ks < 8; ++ks) {
      const int kb = ks * 4 + hi * 2;
      v2f af = { aS[m * 32 + kb], aS[m * 32 + kb + 1] };
      pa[0] = wmma_f32_k4(af, amf[ks][0], pa[0]);
      pa[1] = wmma_f32_k4(af, amf[ks][1], pa[1]);
    }
    #pragma unroll
    for (int nt = 0; nt < 2; ++nt)
      #pragma unroll
      for (int v = 0; v < 8; ++v)
        AS[(v + 8 * hi) * 32 + nt * 16 + m] = pa[nt][v];

    // --- AX(32x32 cplx) += A^T(32x16) x X(16x32), X from sincos ------------
    #pragma unroll
    for (int ks = 0; ks < 4; ++ks) {
      const int kb = ks * 4 + hi * 2;
      const float x00 = xS[kb * 4 + 0], x01 = xS[kb * 4 + 1],
                  x02 = xS[kb * 4 + 2];
      const float x10 = xS[(kb + 1) * 4 + 0], x11 = xS[(kb + 1) * 4 + 1],
                  x12 = xS[(kb + 1) * 4 + 2];
      v2f aT0 = { AS[kb * 32 + m],      AS[(kb + 1) * 32 + m] };
      v2f aT1 = { AS[kb * 32 + 16 + m], AS[(kb + 1) * 32 + 16 + m] };
      #pragma unroll
      for (int gt = 0; gt < 2; ++gt) {
        const float ph0 = TWO_PI * (x00 * xmg[gt][0] + x01 * xmg[gt][1] +
                                    x02 * xmg[gt][2]);
        const float ph1 = TWO_PI * (x10 * xmg[gt][0] + x11 * xmg[gt][1] +
                                    x12 * xmg[gt][2]);
        float s0, c0, s1, c1;
        __sincosf(ph0, &s0, &c0);
        __sincosf(ph1, &s1, &c1);
        v2f bRe = {  c0,  c1 };              // X = exp(-2pi i ph)
        v2f bIm = { -s0, -s1 };
        accRe[0][gt] = wmma_f32_k4(aT0, bRe, accRe[0][gt]);
        accRe[1][gt] = wmma_f32_k4(aT1, bRe, accRe[1][gt]);
        accIm[0][gt] = wmma_f32_k4(aT0, bIm, accIm[0][gt]);
        accIm[1][gt] = wmma_f32_k4(aT1, bIm, accIm[1][gt]);
      }
    }
  }

  // Flush partial AX with HW fp32 atomics (global_atomic_add_f32).
  float* axb = axbuf + (size_t)((batch * N_HEADS + h) * 2) * 1024;
  #pragma unroll
  for (int ft = 0; ft < 2; ++ft)
    #pragma unroll
    for (int gt = 0; gt < 2; ++gt)
      #pragma unroll
      for (int v = 0; v < 8; ++v) {
        const int f = ft * 16 + v + 8 * hi;
        const int g = gt * 16 + m;
        unsafeAtomicAdd(axb + f * 32 + g,        accRe[ft][gt][v]);
        unsafeAtomicAdd(axb + 1024 + f * 32 + g, accIm[ft][gt][v]);
      }
}

// ---------------------------------------------------------------------------
// Kernel 2: U = Re(AX · exp(+2pi i y·y_modes))/S ; out = gelu(U·u_modes +
//           a·fc_w^T + fc_b).  Each wave owns 16 points, loops heads; no
//           cross-wave barriers in the main loop.
// ---------------------------------------------------------------------------
#define K2_BLOCKS_PER_BATCH 216
#define K2_ITERS            8       // 64 points per block-iter (16/wave)

__global__ void __launch_bounds__(128)
k2_finalize(const float* __restrict__ a,
            const float* __restrict__ y,
            const float* __restrict__ axbuf,
            const float* __restrict__ u_modes,
            const float* __restrict__ fc_w,
            const float* __restrict__ fc_b,
            const float* __restrict__ y_modes,
            float* __restrict__ out)
{
  extern __shared__ float smem[];
  float* axRe = smem;                // [4][32][32]: axRe[h][g][f] = AXre[h][f][g]
  float* axIm = axRe + 4096;         // [4][32][32]
  float* um   = axIm + 4096;         // [4*32][32] u_modes[hf][o] * (1/NPTS)
  float* fcwT = um + 4096;           // [32][32]  fcwT[i][o] = fc_w[o][i]
  float* fcb  = fcwT + 1024;         // [32]
  float* ym   = fcb + 32;            // [4][32][3]

  const int tid  = threadIdx.x;
  const int wave = tid >> 5;
  const int lane = tid & 31;
  const int m    = lane & 15;
  const int hi   = lane >> 4;
  const int batch = blockIdx.x / K2_BLOCKS_PER_BATCH;
  const int blk   = blockIdx.x % K2_BLOCKS_PER_BATCH;

  float* wv = ym + 384 + wave * (512 + 2048);
  float* aS = wv;                    // [16][32]
  float* US = aS + 512;              // [16][128] U[p][h*32+f]

  const float invS = 1.0f / (float)NPTS;

  for (int idx = tid; idx < 8192; idx += blockDim.x) {
    int reim = idx >> 12, r = idx & 4095;
    int h = r >> 10, q = r & 1023, f = q >> 5, g = q & 31;
    float val = axbuf[(size_t)(((batch * N_HEADS + h) * 2 + reim) << 10) + f * 32 + g];
    (reim ? axIm : axRe)[(h << 10) + g * 32 + f] = val;   // transpose to [g][f]
  }
  for (int idx = tid; idx < 4096; idx += blockDim.x)
    um[idx] = u_modes[idx] * invS;                        // fold 1/n_xyz
  for (int idx = tid; idx < 1024; idx += blockDim.x) {
    int o = idx >> 5, i = idx & 31;
    fcwT[i * 32 + o] = fc_w[idx];
  }
  for (int idx = tid; idx < 32;  idx += blockDim.x) fcb[idx] = fc_b[idx];
  for (int idx = tid; idx < 384; idx += blockDim.x) ym[idx]  = y_modes[idx];
  __syncthreads();

  // ---- Hoist loop-invariant fragments into registers --------------------
  v2f wf[8][2];                      // fc_w^T B-fragments (skip path)
  #pragma unroll
  for (int ks = 0; ks < 8; ++ks) {
    const int kb = ks * 4 + hi * 2;
    #pragma unroll
    for (int ot = 0; ot < 2; ++ot) {
      wf[ks][ot][0] = fcwT[kb * 32 + ot * 16 + m];
      wf[ks][ot][1] = fcwT[(kb + 1) * 32 + ot * 16 + m];
    }
  }
  const float bias0 = fcb[m];
  const float bias1 = fcb[16 + m];

  const v8f vz = {0.f,0.f,0.f,0.f,0.f,0.f,0.f,0.f};

  for (int it = 0; it < K2_ITERS; ++it) {
    const long p0 = (long)batch * NPTS + (long)blk * (K2_ITERS * 64)
                  + it * 64 + wave * 16;

    for (int r = 0; r < 16; ++r)
      aS[r * 32 + lane] = a[(p0 + r) * C_IN + lane];
    const float yv0 = y[(p0 + m) * 3 + 0];   // this lane's point row (M=m)
    const float yv1 = y[(p0 + m) * 3 + 1];
    const float yv2 = y[(p0 + m) * 3 + 2];

    // --- U(16x32) per head: A-frag = Y phases on the fly, B-frag = AX^T ----
    for (int h = 0; h < N_HEADS; ++h) {
      v8f uacc[2] = {vz, vz};
      #pragma unroll
      for (int ks = 0; ks < 8; ++ks) {
        const int kb = ks * 4 + hi * 2;
        const float* ymh = ym + h * 96;
        float ph0 = TWO_PI * (yv0 * ymh[kb * 3 + 0] +
                              yv1 * ymh[kb * 3 + 1] +
                              yv2 * ymh[kb * 3 + 2]);
        float ph1 = TWO_PI * (yv0 * ymh[(kb + 1) * 3 + 0] +
                              yv1 * ymh[(kb + 1) * 3 + 1] +
                              yv2 * ymh[(kb + 1) * 3 + 2]);
        float s0, c0, s1, c1;
        __sincosf(ph0, &s0, &c0);
        __sincosf(ph1, &s1, &c1);
        v2f aRe = {  c0,  c1 };              // Re(AX)*Re(Y)
        v2f aIm = { -s0, -s1 };              // -Im(Y)*Im(AX)
        #pragma unroll
        for (int ft = 0; ft < 2; ++ft) {
          v2f bRe = { axRe[(h << 10) + kb * 32 + ft * 16 + m],
                      axRe[(h << 10) + (kb + 1) * 32 + ft * 16 + m] };
          v2f bIm = { axIm[(h << 10) + kb * 32 + ft * 16 + m],
                      axIm[(h << 10) + (kb + 1) * 32 + ft * 16 + m] };
          uacc[ft] = wmma_f32_k4(aRe, bRe, uacc[ft]);
          uacc[ft] = wmma_f32_k4(aIm, bIm, uacc[ft]);
        }
      }
      #pragma unroll
      for (int ft = 0; ft < 2; ++ft)
        #pragma unroll
        for (int v = 0; v < 8; ++v)
          US[(v + 8 * hi) * 128 + h * 32 + ft * 16 + m] = uacc[ft][v];
    }

    // --- out(16x32) = gelu( US·um (K=128) + aS·fcwT (K=32) + fcb ) ---------
    v8f oacc[2];
    #pragma unroll
    for (int v = 0; v < 8; ++v) { oacc[0][v] = bias0; oacc[1][v] = bias1; }

    #pragma unroll
    for (int ks = 0; ks < 8; ++ks) {                     // skip path, K = C_IN
      const int kb = ks * 4 + hi * 2;
      v2f af = { aS[m * 32 + kb], aS[m * 32 + kb + 1] };
      oacc[0] = wmma_f32_k4(af, wf[ks][0], oacc[0]);
      oacc[1] = wmma_f32_k4(af, wf[ks][1], oacc[1]);
    }
    #pragma unroll
    for (int ks = 0; ks < 32; ++ks) {                    // attention, K = h*f
      const int kb = ks * 4 + hi * 2;
      v2f af = { US[m * 128 + kb], US[m * 128 + kb + 1] };
      #pragma unroll
      for (int ot = 0; ot < 2; ++ot) {
        v2f bf = { um[kb * 32 + ot * 16 + m],
                   um[(kb + 1) * 32 + ot * 16 + m] };
        oacc[ot] = wmma_f32_k4(af, bf, oacc[ot]);
      }
    }
    #pragma unroll
    for (int ot = 0; ot < 2; ++ot)
      #pragma unroll
      for (int v = 0; v < 8; ++v) {
        const float val = oacc[ot][v];
        const float gel = 0.5f * val *
            (1.0f + erff(val * 0.70710678118654752440f));   // exact GELU
        out[(p0 + v + 8 * hi) * C_OUT + ot * 16 + m] = gel;
      }
  }
}

// ---------------------------------------------------------------------------
extern "C" void kernel_launch(void* const* d_in, const int* in_sizes, int n_in,
                              void* d_out, int out_size, void* d_ws,
                              size_t ws_size, hipStream_t stream) {
  const float* a       = (const float*)d_in[0];
  const float* x       = (const float*)d_in[1];
  const float* y       = (const float*)d_in[2];
  const float* a_modes = (const float*)d_in[3];
  const float* x_modes = (const float*)d_in[4];
  const float* y_modes = (const float*)d_in[5];
  const float* u_modes = (const float*)d_in[6];
  const float* fc_w    = (const float*)d_in[7];
  const float* fc_b    = (const float*)d_in[8];
  float* out   = (float*)d_out;
  float* axbuf = (float*)d_ws;       // [2][4][2][32][32] f32 = 64 KB

  (void)in_sizes; (void)n_in; (void)out_size; (void)ws_size;

  hipMemsetAsync(axbuf, 0, (size_t)NBATCH * N_HEADS * 2 * 1024 * sizeof(float),
                 stream);

  const size_t lds1 = (size_t)(4096 + 384 + 4 * (512 + 64 + 512)) * sizeof(float);
  k1_accumulate_AX<<<dim3(NBATCH * K1_BLOCKS_PER_BATCH), dim3(128), lds1,
                     stream>>>(a, x, a_modes, x_modes, axbuf);

  const size_t lds2 = (size_t)(4096 * 3 + 1024 + 32 + 384 + 4 * (512 + 2048)) *
                      sizeof(float);
  k2_finalize<<<dim3(NBATCH * K2_BLOCKS_PER_BATCH), dim3(128), lds2, stream>>>(
      a, y, axbuf, u_modes, fc_w, fc_b, y_modes, out);
}